// GraphSAGEDecoder_29180007809576
// MI455X (gfx1250) — compile-verified
//
#include <hip/hip_runtime.h>
#include <hip/hip_bf16.h>
#include <stdint.h>

// ---------------------------------------------------------------------------
// 3-layer GCN: h = dropout(leakyrelu( (D_out^-1/2 A D_in^-1/2) (X*ns) W + b ))
// fp32 throughout (matches reference); GEMM uses V_WMMA_F32_16X16X4_F32 with
// async-to-LDS A staging and register-resident B fragments.
// ---------------------------------------------------------------------------

typedef __attribute__((ext_vector_type(2))) float v2f;
typedef __attribute__((ext_vector_type(8))) float v8f;
typedef __attribute__((ext_vector_type(4))) int   v4i;
typedef v4i __attribute__((address_space(1))) gv4i;  // global int4
typedef v4i __attribute__((address_space(3))) sv4i;  // LDS int4

#define FIN 128
#define LDA 132  // padded LDS row stride (floats): 132 % 64 = 4 -> no bank conflicts
#define LEAKY 0.01f

// ---- deterministic stateless uniform hash (substitute for JAX threefry) ----
__device__ __forceinline__ float u01_hash(uint32_t a, uint32_t b, uint32_t c) {
  uint32_t x = a * 0x9E3779B1u;
  x ^= b + 0x85EBCA77u + (x << 6) + (x >> 2);
  x ^= c + 0xC2B2AE3Du + (x << 6) + (x >> 2);
  x ^= x >> 16; x *= 0x7FEB352Du;
  x ^= x >> 15; x *= 0x846CA68Bu;
  x ^= x >> 16;
  return (float)(x >> 8) * (1.0f / 16777216.0f);
}

// ---- utility: zero a float buffer ------------------------------------------
__global__ void zero_f32(float* __restrict__ p, size_t count) {
  size_t i = (size_t)blockIdx.x * blockDim.x + threadIdx.x;
  size_t stride = (size_t)gridDim.x * blockDim.x;
  for (; i < count; i += stride) p[i] = 0.0f;
}

// ---- degree accumulation (fp32 atomics; counts exact up to 2^24) -----------
__global__ void degree_kernel(const int* __restrict__ src, const int* __restrict__ dst,
                              float* __restrict__ deg_out, float* __restrict__ deg_in,
                              int n_edges) {
  int i = blockIdx.x * blockDim.x + threadIdx.x;
  int stride = gridDim.x * blockDim.x;
  for (; i < n_edges; i += stride) {
    atomicAdd(&deg_out[src[i]], 1.0f);
    atomicAdd(&deg_in[dst[i]], 1.0f);
  }
}

__global__ void norm_finalize(float* __restrict__ p, int n) {
  int i = blockIdx.x * blockDim.x + threadIdx.x;
  if (i < n) p[i] = rsqrtf(fmaxf(p[i], 1.0f));
}

// ---- GEMM: H[n x FOUT] = (X[n x 128] * norm_src[:,None]) @ W[128 x FOUT] ----
// 64-row M supertile per block. Wave w owns N-tile [16w,16w+16) and keeps its
// whole B fragment set (FIN/4 x v2f = 64 VGPRs) in registers, reused across
// 4 M tiles (128 WMMAs/wave). A staged raw via async global->LDS DMA; the
// norm_src row scaling is applied on the A fragment (co-executes with XDL).
template <int FOUT>
__global__ __launch_bounds__(32 * (FOUT / 16))
void gemm_norm_wmma(const float* __restrict__ X, const float* __restrict__ norm_src,
                    const float* __restrict__ W, float* __restrict__ H, int n) {
  __shared__ float Atile[64 * LDA];  // 33 KB, padded rows

  const int row0 = blockIdx.x * 64;
  const int tid = threadIdx.x;
  const int nthreads = 32 * (FOUT / 16);
  const bool full = (row0 + 64 <= n);

  if (full) {
    // async DMA: 16B per issue, 64*128/4 = 2048 chunks; ASYNCcnt-tracked.
    for (int i = tid; i < 64 * (FIN / 4); i += nthreads) {
      const int r = i >> 5;            // chunk row   (32 chunks per row)
      const int c = (i & 31) * 4;      // chunk col
      const float* g = X + (size_t)(row0 + r) * FIN + c;
      __builtin_amdgcn_global_load_async_to_lds_b128(
          (gv4i*)g, (sv4i*)&Atile[r * LDA + c], 0, 0);
    }
    asm volatile("s_wait_asynccnt 0" ::: "memory");
  } else {
    for (int i = tid; i < 64 * FIN; i += nthreads) {
      const int r = i >> 7;
      const int c = i & (FIN - 1);
      const int row = row0 + r;
      Atile[r * LDA + c] = (row < n) ? X[(size_t)row * FIN + c] : 0.0f;
    }
  }
  __syncthreads();

  const int wave = tid >> 5;
  const int lane = tid & 31;
  const int ncol = wave * 16 + (lane & 15);
  const int kb = (lane >> 4) * 2;     // fp32 16x16x4: lanes 16-31 hold K=2,3
  const int m = lane & 15;
  const int mbase = (lane >> 4) * 8;  // D layout: vgpr r -> row r + 8*(l>>4)

  // preload B fragments once (W <= 64KB, L2/WGP$ resident)
  v2f bfrag[FIN / 4];
#pragma unroll
  for (int kk = 0; kk < FIN / 4; ++kk) {
    const int k = kk * 4 + kb;
    bfrag[kk].x = W[(size_t)k * FOUT + ncol];
    bfrag[kk].y = W[(size_t)(k + 1) * FOUT + ncol];
  }

#pragma unroll
  for (int mt = 0; mt < 4; ++mt) {
    const int rowm = row0 + mt * 16 + m;
    float ns = 0.0f;
    if (rowm < n) ns = norm_src[rowm];
    v8f acc = {};
#pragma unroll
    for (int kk = 0; kk < FIN / 4; ++kk) {
      v2f a = *(const v2f*)&Atile[(mt * 16 + m) * LDA + kk * 4 + kb];
      a *= ns;  // row scaling fused into fragment
      acc = __builtin_amdgcn_wmma_f32_16x16x4_f32(
          /*neg_a=*/false, a, /*neg_b=*/false, bfrag[kk],
          /*c_mod=*/(short)0, acc, /*reuse_a=*/false, /*reuse_b=*/false);
    }
    const int orow0 = row0 + mt * 16 + mbase;
    if (full) {  // uniform fast path: guard-free clause of 8 stores
#pragma unroll
      for (int r = 0; r < 8; ++r)
        H[(size_t)(orow0 + r) * FOUT + ncol] = acc[r];
    } else {
#pragma unroll
      for (int r = 0; r < 8; ++r) {
        const int row = orow0 + r;
        if (row < n) H[(size_t)row * FOUT + ncol] = acc[r];
      }
    }
  }
}

// ---- SpMM scatter: agg[dst[e]] += H[src[e]]  (edge x feature-chunk) --------
// float4 gathers (coalesced per-edge), 4x global_atomic_add_f32 scatters.
// Working set fits the 192MB L2, so atomics resolve in L2, not HBM.
template <int F>
__global__ void spmm_scatter(const int* __restrict__ src, const int* __restrict__ dst,
                             const float* __restrict__ H, float* __restrict__ agg,
                             int n_edges) {
  const int CH = F / 4;
  const long long total = (long long)n_edges * CH;
  long long idx = (long long)blockIdx.x * blockDim.x + threadIdx.x;
  const long long stride = (long long)gridDim.x * blockDim.x;
  for (; idx < total; idx += stride) {
    const int e = (int)(idx / CH);
    const int c = (int)(idx % CH);
    const int s = src[e];
    const int d = dst[e];
    const float4 v = *(const float4*)(H + (size_t)s * F + c * 4);
    float* o = agg + (size_t)d * F + c * 4;
    atomicAdd(o + 0, v.x);
    atomicAdd(o + 1, v.y);
    atomicAdd(o + 2, v.z);
    atomicAdd(o + 3, v.w);
  }
}

// ---- epilogue: out = dropout(leakyrelu(agg*norm_dst + b)); in-place safe ---
template <int F>
__global__ void epilogue_kernel(const float* __restrict__ agg, const float* __restrict__ norm_dst,
                                const float* __restrict__ bias, float* __restrict__ out,
                                int n, uint32_t seed) {
  const long long total = (long long)n * F;
  long long idx = (long long)blockIdx.x * blockDim.x + threadIdx.x;
  const long long stride = (long long)gridDim.x * blockDim.x;
  for (; idx < total; idx += stride) {
    const int node = (int)(idx / F);
    const int f = (int)(idx % F);
    float v = agg[idx] * norm_dst[node] + bias[f];
    v = (v > 0.0f) ? v : LEAKY * v;
    const float u = u01_hash(seed, (uint32_t)node, (uint32_t)f);
    out[idx] = (u >= 0.5f) ? v * 2.0f : 0.0f;  // p=0.5, keep-scale 1/(1-p)
  }
}

// ---------------------------------------------------------------------------
extern "C" void kernel_launch(void* const* d_in, const int* in_sizes, int n_in,
                              void* d_out, int out_size, void* d_ws, size_t ws_size,
                              hipStream_t stream) {
  const float* features = (const float*)d_in[0];
  const int*   src      = (const int*)d_in[1];
  const int*   dst      = (const int*)d_in[2];
  const float* W1 = (const float*)d_in[3];
  const float* b1 = (const float*)d_in[4];
  const float* W2 = (const float*)d_in[5];
  const float* b2 = (const float*)d_in[6];
  const float* W3 = (const float*)d_in[7];
  const float* b3 = (const float*)d_in[8];

  const int n = in_sizes[0] / FIN;   // 50000
  const int e = in_sizes[1];         // 800000
  float* out = (float*)d_out;

  // workspace layout (floats)
  float* ws = (float*)d_ws;
  float* norm_src = ws;                 ws += n;
  float* norm_dst = ws;                 ws += n;
  float* bufA = ws;                     ws += (size_t)n * FIN;
  float* bufB = ws;                     ws += (size_t)n * FIN;
  float* bufC = ws;                     ws += (size_t)n * FIN;

  const int T = 256;
  const int zb_n   = (int)(((size_t)n + T - 1) / T);
  const int zb_nf  = (int)(((size_t)n * FIN + T - 1) / T);
  const int eb     = (e + T - 1) / T;
  const int gemm_blocks = (n + 63) / 64;

  // ---- degrees -> rsqrt norms ----
  zero_f32<<<zb_n, T, 0, stream>>>(norm_src, (size_t)n);
  zero_f32<<<zb_n, T, 0, stream>>>(norm_dst, (size_t)n);
  degree_kernel<<<eb, T, 0, stream>>>(src, dst, norm_src, norm_dst, e);
  norm_finalize<<<zb_n, T, 0, stream>>>(norm_src, n);
  norm_finalize<<<zb_n, T, 0, stream>>>(norm_dst, n);

  const long long sp128 = (long long)e * (128 / 4);
  const int sp128_b = (int)((sp128 + T - 1) / T);
  const long long sp64 = (long long)e * (64 / 4);
  const int sp64_b = (int)((sp64 + T - 1) / T);
  const int ep128_b = (int)(((long long)n * 128 + T - 1) / T);
  const int ep64_b  = (int)(((long long)n * 64 + T - 1) / T);

  // ---- layer 1: features -> bufB ----
  gemm_norm_wmma<128><<<gemm_blocks, 256, 0, stream>>>(features, norm_src, W1, bufA, n);
  zero_f32<<<zb_nf, T, 0, stream>>>(bufB, (size_t)n * 128);
  spmm_scatter<128><<<sp128_b, T, 0, stream>>>(src, dst, bufA, bufB, e);
  epilogue_kernel<128><<<ep128_b, T, 0, stream>>>(bufB, norm_dst, b1, bufB, n, 0x42000001u);

  // ---- layer 2: bufB -> bufC ----
  gemm_norm_wmma<128><<<gemm_blocks, 256, 0, stream>>>(bufB, norm_src, W2, bufA, n);
  zero_f32<<<zb_nf, T, 0, stream>>>(bufC, (size_t)n * 128);
  spmm_scatter<128><<<sp128_b, T, 0, stream>>>(src, dst, bufA, bufC, e);
  epilogue_kernel<128><<<ep128_b, T, 0, stream>>>(bufC, norm_dst, b2, bufC, n, 0x42000002u);

  // ---- layer 3: bufC -> d_out (F = 64) ----
  gemm_norm_wmma<64><<<gemm_blocks, 128, 0, stream>>>(bufC, norm_src, W3, bufA, n);
  zero_f32<<<ep64_b, T, 0, stream>>>(bufB, (size_t)n * 64);
  spmm_scatter<64><<<sp64_b, T, 0, stream>>>(src, dst, bufA, bufB, e);
  epilogue_kernel<64><<<ep64_b, T, 0, stream>>>(bufB, norm_dst, b3, out, n, 0x42000003u);
}